// WaveletTransform_31516470018523
// MI455X (gfx1250) — compile-verified
//
#include <hip/hip_runtime.h>
#include <hip/hip_bf16.h>

typedef float v2f __attribute__((ext_vector_type(2)));
typedef float v8f __attribute__((ext_vector_type(8)));

// K_c[k] = filt_c[7-k]  (pre-flipped db4 dec_lo / dec_hi, pywt convention)
__constant__ float c_kern[2][8] = {
  {  0.23037781330885523f,  0.7148465705525415f,   0.6308807679295904f,  -0.02798376941698385f,
    -0.18703481171888114f,  0.030841381835986965f, 0.032883011666982945f,-0.010597401784997278f },
  { -0.010597401784997278f,-0.032883011666982945f, 0.030841381835986965f, 0.18703481171888114f,
    -0.02798376941698385f, -0.6308807679295904f,   0.7148465705525415f,  -0.23037781330885523f }
};

// Branchless pywt symmetric (half-sample) extension map for ext[] (len n+14):
//   left reflect:  6-pos  == ~(pos-7)        when pos-7 < 0
//   right reflect: 2n+6-pos == 2n-1-(pos-7)  when pos-7 >= n
// plus clamp for safe over-reads on the last partial tile.
__device__ __forceinline__ int ext_map(int pos, int n) {
  int s = pos - 7;
  s = (s < 0) ? ~s : s;
  int r = 2 * n - 1 - s;
  s = (s >= n) ? r : s;
  return min(max(s, 0), n - 1);
}

// One DWT level.  Signal layout: [b][i][64 features] (f contiguous -> coalesced).
// Each wave handles 16 features of one batch over a segment of 8-output tiles:
//   D(16x16) = sum_{j=0..5} A_j(16x4) * B_j(4x16)     (6 chained f32 WMMAs)
// A rows 0-7 = low-pass Toeplitz, rows 8-15 = high-pass; B cols = features.
// D lanes 0-15 carry cA outputs, lanes 16-31 carry cD outputs.
template<int STORECA, int ACCCA>
__global__ __launch_bounds__(256)
void dwt_level_kernel(const float* __restrict__ in, float* __restrict__ caOut,
                      float* __restrict__ pCD, float* __restrict__ pCA,
                      int Lin, int Lout, int segs, int tilesPerSeg)
{
  const int lane = threadIdx.x & 31;
  const int wid  = (int)((blockIdx.x * blockDim.x + threadIdx.x) >> 5);
  const int seg  = wid % segs;
  const int fg   = (wid / segs) & 3;
  const int b    = wid / (segs * 4);

  const int nf   = lane & 15;        // feature within group (B column) and A row
  const int kb   = (lane >> 4) * 2;  // K-pair base for this half-wave
  const int f    = fg * 16 + nf;
  const int mm   = nf & 7;           // A row m & 7 (output position in tile)
  const int band = nf >> 3;          // 0 = low-pass rows, 1 = high-pass rows

  // Per-lane constant A fragments: A[m][q] = K_band[q - 2*(m&7)] (Toeplitz)
  v2f a[6];
#pragma unroll
  for (int j = 0; j < 6; ++j) {
    int q0 = 4 * j + kb;
    int t0 = q0 - 2 * mm;
    int t1 = t0 + 1;
    a[j].x = (t0 >= 0 && t0 < 8) ? c_kern[band][t0] : 0.0f;
    a[j].y = (t1 >= 0 && t1 < 8) ? c_kern[band][t1] : 0.0f;
  }

  // Wave-uniform bases + 32-bit per-lane offsets for the boundary path.
  const float* baseu = in + (size_t)b * (size_t)Lin * 64;   // uniform across wave
  const float* sig   = baseu + f;                           // per-lane, fast path
  float* caOu = STORECA ? (caOut + (size_t)b * (size_t)Lout * 64) : nullptr; // uniform
  float* caO  = STORECA ? (caOu + f) : nullptr;                              // per-lane

  const int tiles = (Lout + 7) >> 3;
  const int tBeg  = seg * tilesPerSeg;
  const int tEnd  = min(tBeg + tilesPerSeg, tiles);

  // Fast region: tile>=1 (no left reflection), window fits (no right
  // reflection), and all 8 outputs valid.
  const int fastLim = min((Lin - 18) >> 4, (Lout - 8) >> 3);  // last fast tile
  const int hEnd    = min(tEnd, 1);                 // head slow tiles: [tBeg,hEnd)
  const int fBeg    = min(max(tBeg, 1), tEnd);      // fast tiles:      [fBeg,fEnd)
  const int fEnd    = min(tEnd, fastLim + 1);
  const int tailBeg = max(fBeg, fEnd);              // tail slow tiles: [tailBeg,tEnd)

  float sSum = 0.f, sSq = 0.f, sAbs = 0.f, sLg = 0.f;

  auto accum = [&](float d) {
    float ad = fabsf(d);
    sSum += d; sSq += d * d; sAbs += ad;
    sLg  += ad * __log2f(ad + 1e-30f);
  };

  // Which lanes accumulate stats: cD lanes always; cA lanes only when ACCCA.
  const bool accLane = ACCCA ? true : (lane >= 16);
  const bool loLane  = (lane < 16);

  // Boundary tiles (head + tail) in one loop so the body inlines exactly once.
#pragma unroll 1
  for (int phase = 0; phase < 2; ++phase) {
    const int lo = phase ? tailBeg : tBeg;
    const int hi = phase ? tEnd    : hEnd;
#pragma unroll 1
    for (int tile = lo; tile < hi; ++tile) {
      const int t0o  = tile * 8;
      const int base = 2 * t0o + 1;
      v8f c = {};
#pragma unroll
      for (int j = 0; j < 6; ++j) {
        int p0 = base + 4 * j + kb;
        v2f bv;
        bv.x = baseu[(unsigned)(ext_map(p0,     Lin) * 64 + f)];
        bv.y = baseu[(unsigned)(ext_map(p0 + 1, Lin) * 64 + f)];
        c = __builtin_amdgcn_wmma_f32_16x16x4_f32(false, a[j], false, bv,
                                                  (short)0, c, false, false);
      }
#pragma unroll
      for (int v = 0; v < 8; ++v) {
        const int  i     = t0o + v;
        const bool valid = (i < Lout);
        const float d    = c[v];
        // Branchless gated accumulation: a zero contributes nothing
        // (0*log2(1e-30) == 0), so no exec manipulation needed.
        accum((valid && accLane) ? d : 0.0f);
        if (STORECA) {
          if (valid && loLane) caOu[(unsigned)(i * 64 + f)] = d;
        }
      }
    }
  }

  // ---- fast interior tiles: immediate-offset loads, no predication ----
  {
    // ext index base-7 = 16*tile - 6 ; fold kb into the pointer.
    const float* pf = sig + (size_t)(16 * fBeg - 6 + kb) * 64;
    float*       qf = STORECA ? (caO + (size_t)fBeg * 8 * 64) : nullptr;
#pragma unroll 1
    for (int tile = fBeg; tile < fEnd; ++tile) {
      __builtin_prefetch((const void*)(pf + 16 * 64), 0, 1);  // next tile
      v2f bv[6];
#pragma unroll
      for (int j = 0; j < 6; ++j) {        // byte offsets: j*1024 and +256
        bv[j].x = pf[j * 256];
        bv[j].y = pf[j * 256 + 64];
      }
      v8f c = {};
#pragma unroll
      for (int j = 0; j < 6; ++j)
        c = __builtin_amdgcn_wmma_f32_16x16x4_f32(false, a[j], false, bv[j],
                                                  (short)0, c, false, false);
      if (lane < 16) {
        if (STORECA) {
#pragma unroll
          for (int v = 0; v < 8; ++v) qf[v * 64] = c[v];
        }
        if (ACCCA) {
#pragma unroll
          for (int v = 0; v < 8; ++v) accum(c[v]);
        }
      } else {
#pragma unroll
        for (int v = 0; v < 8; ++v) accum(c[v]);
      }
      pf += 16 * 64;
      if (STORECA) qf += 8 * 64;
    }
  }

  // Segment partials: [seg][b*64+f] as float4 -> global_store_b128
  const size_t sidx = (size_t)seg * 4096 + (size_t)b * 64 + f;
  const float4 part = make_float4(sSum, sSq, sAbs, sLg);
  if (lane >= 16) {
    ((float4*)pCD)[sidx] = part;
  } else if (ACCCA) {
    ((float4*)pCA)[sidx] = part;
  }
}

// Deterministic fixed-order reduction of segment partials + stat finalize.
// out[b, f*16 + band*4 + stat], bands = [cA3, cD3, cD2, cD1].
__global__ __launch_bounds__(256)
void finalize_kernel(const float* __restrict__ pA3, const float* __restrict__ pD3,
                     const float* __restrict__ pD2, const float* __restrict__ pD1,
                     float* __restrict__ out)
{
  const int t = blockIdx.x * blockDim.x + threadIdx.x;
  if (t >= 4096) return;
  const int b = t >> 6, f = t & 63;

  const float* parts[4] = { pA3, pD3, pD2, pD1 };
  const int   segsArr[4] = { 4, 4, 8, 16 };
  const float lenArr[4]  = { 1030.f, 1030.f, 2053.f, 4099.f };

  for (int band = 0; band < 4; ++band) {
    float sSum = 0.f, sSq = 0.f, sAbs = 0.f, sLg = 0.f;
    const float4* p = (const float4*)parts[band];
    const int S = segsArr[band];
    for (int s = 0; s < S; ++s) {
      float4 v = p[(size_t)s * 4096 + t];
      sSum += v.x; sSq += v.y; sAbs += v.z; sLg += v.w;
    }
    const float L    = lenArr[band];
    const float mean = sSum / L;
    const float var  = fmaxf(sSq / L - mean * mean, 0.f);
    const float sd   = sqrtf(var);
    const float ent  = (sAbs > 0.f) ? (__log2f(sAbs) - sLg / sAbs) : 0.f;
    const size_t o   = (size_t)b * 1024 + (size_t)f * 16 + (size_t)band * 4;
    out[o + 0] = mean; out[o + 1] = sd; out[o + 2] = sSq; out[o + 3] = ent;
  }
}

extern "C" void kernel_launch(void* const* d_in, const int* in_sizes, int n_in,
                              void* d_out, int out_size, void* d_ws, size_t ws_size,
                              hipStream_t stream)
{
  const float* x = (const float*)d_in[0];
  float* ws  = (float*)d_ws;

  // Workspace layout (floats):
  float* cA1 = ws;                                   // 64*4099*64 = 16,789,504
  float* cA2 = cA1 + (size_t)64 * 4099 * 64;         // 64*2053*64 =  8,409,088
  float* pD1 = cA2 + (size_t)64 * 2053 * 64;         // 16*4096*4
  float* pD2 = pD1 + (size_t)16 * 4096 * 4;          //  8*4096*4
  float* pD3 = pD2 + (size_t)8  * 4096 * 4;          //  4*4096*4
  float* pA3 = pD3 + (size_t)4  * 4096 * 4;          //  4*4096*4
  (void)ws_size; (void)in_sizes; (void)n_in; (void)out_size;

  // Level 1: 8192 -> 4099 ; 64 b * 4 fgroups * 16 segs = 4096 waves
  dwt_level_kernel<1, 0><<<512, 256, 0, stream>>>(x,   cA1, pD1, nullptr, 8192, 4099, 16, 33);
  // Level 2: 4099 -> 2053 ; 2048 waves
  dwt_level_kernel<1, 0><<<256, 256, 0, stream>>>(cA1, cA2, pD2, nullptr, 4099, 2053,  8, 33);
  // Level 3: 2053 -> 1030 ; stats only (cA3 + cD3); 1024 waves
  dwt_level_kernel<0, 1><<<128, 256, 0, stream>>>(cA2, nullptr, pD3, pA3, 2053, 1030,  4, 33);
  // Reduce partials and emit [64, 1024] stats
  finalize_kernel<<<16, 256, 0, stream>>>(pA3, pD3, pD2, pD1, (float*)d_out);
}